// Fourier_layer_88596585382623
// MI455X (gfx1250) — compile-verified
//
#include <hip/hip_runtime.h>
#include <hip/hip_bf16.h>

typedef float v2f __attribute__((ext_vector_type(2)));
typedef float v8f __attribute__((ext_vector_type(8)));
typedef int gv4i __attribute__((vector_size(16)));   // matches builtin param

#define SS 31     // S (modes per axis)
#define P  62     // 2*S : fused pair-index extent
#define PD 64     // padded to 4 WMMA tiles
#define NB 8      // batch
#define ST 68     // LDS row stride in floats (272B: 16B-aligned, bank-friendly)

#define AS1 __attribute__((address_space(1)))
#define AS3 __attribute__((address_space(3)))

#define WMMA_F32(a, b, c)                                                     \
  __builtin_amdgcn_wmma_f32_16x16x4_f32(false, (a), false, (b), (short)0,     \
                                        (c), false, false)

#if __has_builtin(__builtin_amdgcn_global_load_async_to_lds_b128)
#define HAVE_ASYNC_LDS 1
#else
#define HAVE_ASYNC_LDS 0
#endif

// ---------------------------------------------------------------------------
// Stage one 64x64 fp32 matrix (row-major, packed) into LDS with padded row
// stride ST, in 16-byte chunks. 128 threads -> 8 chunks per thread.
// ---------------------------------------------------------------------------
__device__ __forceinline__ void stage_matrix(float* lds, const float* g,
                                             int tid) {
  for (int q = tid; q < PD * 16; q += 128) {
    int row = q >> 4;
    int c   = (q & 15) << 2;  // float offset within row, 4-float chunks
#if HAVE_ASYNC_LDS
    __builtin_amdgcn_global_load_async_to_lds_b128(
        (AS1 gv4i*)(uintptr_t)(g + row * PD + c),
        (AS3 gv4i*)(uintptr_t)(lds + row * ST + c), 0, 0);
#else
    *(float4*)(lds + row * ST + c) = *(const float4*)(g + row * PD + c);
#endif
  }
}

__device__ __forceinline__ void stage_fence() {
#if HAVE_ASYNC_LDS
#if __has_builtin(__builtin_amdgcn_s_wait_asynccnt)
  __builtin_amdgcn_s_wait_asynccnt(0);
#else
  asm volatile("s_wait_asynccnt 0" ::: "memory");
#endif
#endif
  __syncthreads();
}

// ---------------------------------------------------------------------------
// Build shared coefficient tensor  C[K][u][v]  (zero padded to 64^3)
//   C[(k,p),(a,x),(b,y)] = weight[a,b] * sum_l fourier[a,b,k,l]*transform[l,p,x,y]
// ---------------------------------------------------------------------------
__global__ void build_coeff(const float* __restrict__ weight,
                            const float* __restrict__ F,   // (S,S,S,4)
                            const float* __restrict__ T,   // (4,2,2,2)
                            float* __restrict__ Cw) {
  int idx = blockIdx.x * blockDim.x + threadIdx.x;          // [0, 64^3)
  if (idx >= PD * PD * PD) return;
  int v = idx & 63, u = (idx >> 6) & 63, K = idx >> 12;
  float val = 0.0f;
  if (K < P && u < P && v < P) {
    int k = K >> 1, p = K & 1;
    int a = u >> 1, x = u & 1;
    int b = v >> 1, y = v & 1;
    const float* f = F + (((a * SS + b) * SS) + k) * 4;
    float s = 0.0f;
#pragma unroll
    for (int l = 0; l < 4; ++l)
      s += f[l] * T[((l * 2 + p) * 2 + x) * 2 + y];
    val = s * weight[a * SS + b];
  }
  Cw[idx] = val;
}

// ---------------------------------------------------------------------------
// Zero-pad in1/in2 from [B][62][62] to [B][64][64]
// ---------------------------------------------------------------------------
__global__ void pad_inputs(const float* __restrict__ in1,
                           const float* __restrict__ in2,
                           float* __restrict__ in1p,
                           float* __restrict__ in2p) {
  int idx = blockIdx.x * blockDim.x + threadIdx.x;          // [0, 8*64*64)
  if (idx >= NB * PD * PD) return;
  int c = idx & 63, row = (idx >> 6) & 63, beta = idx >> 12;
  float v1 = 0.0f, v2 = 0.0f;
  if (row < P && c < P) {
    int src = (beta * P + row) * P + c;
    v1 = in1[src];
    v2 = in2[src];
  }
  in1p[idx] = v1;
  in2p[idx] = v2;
}

// ---------------------------------------------------------------------------
// GEMM1:  T1[beta,K][r][v] = sum_u in1p[beta][u][r] * Cw[K][u][v]
// 128 threads / 4 waves per block; wave w computes rows 16w..16w+15 x all 64
// columns (4 accumulators, A fragment reused 4x). Operands staged in LDS via
// async global->LDS copies. A is consumed transposed straight from LDS.
// ---------------------------------------------------------------------------
__global__ __launch_bounds__(128) void gemm1(const float* __restrict__ in1p,
                                             const float* __restrict__ Cw,
                                             float* __restrict__ T1) {
  __shared__ float lA[PD * ST];
  __shared__ float lB[PD * ST];
  int mat  = blockIdx.x;                 // beta*64 + K
  int beta = mat >> 6;
  int K    = mat & 63;
  stage_matrix(lA, in1p + beta * (PD * PD), threadIdx.x);  // stored [u][r]
  stage_matrix(lB, Cw + K * (PD * PD), threadIdx.x);       // [u][v]
  stage_fence();

  int wave = threadIdx.x >> 5;
  int lane = threadIdx.x & 31;
  int half = lane >> 4, l16 = lane & 15;
  int m0 = wave * 16;                    // r tile row
  v8f acc0 = {}, acc1 = {}, acc2 = {}, acc3 = {};
#pragma unroll
  for (int k = 0; k < PD; k += 4) {
    int kk = k + half * 2;
    v2f a;                               // A^T fragment: lA[u][r]
    a.x = lA[kk * ST + m0 + l16];
    a.y = lA[(kk + 1) * ST + m0 + l16];
    v2f b0, b1, b2, b3;
    b0.x = lB[kk * ST + 0 + l16];  b0.y = lB[(kk + 1) * ST + 0 + l16];
    b1.x = lB[kk * ST + 16 + l16]; b1.y = lB[(kk + 1) * ST + 16 + l16];
    b2.x = lB[kk * ST + 32 + l16]; b2.y = lB[(kk + 1) * ST + 32 + l16];
    b3.x = lB[kk * ST + 48 + l16]; b3.y = lB[(kk + 1) * ST + 48 + l16];
    acc0 = WMMA_F32(a, b0, acc0);
    acc1 = WMMA_F32(a, b1, acc1);
    acc2 = WMMA_F32(a, b2, acc2);
    acc3 = WMMA_F32(a, b3, acc3);
  }
  float* D = T1 + mat * (PD * PD);
#pragma unroll
  for (int r = 0; r < 8; ++r) {
    int row = (m0 + r + half * 8) * PD + l16;
    D[row + 0]  = acc0[r];
    D[row + 16] = acc1[r];
    D[row + 32] = acc2[r];
    D[row + 48] = acc3[r];
  }
}

// ---------------------------------------------------------------------------
// GEMM2:  T2[beta,K][r][s] = sum_v T1[beta,K][r][v] * in2p[beta][v][s]
// Same blocking; A is row-major so its fragment is a contiguous LDS pair.
// ---------------------------------------------------------------------------
__global__ __launch_bounds__(128) void gemm2(const float* __restrict__ T1,
                                             const float* __restrict__ in2p,
                                             float* __restrict__ T2) {
  __shared__ float lA[PD * ST];
  __shared__ float lB[PD * ST];
  int mat  = blockIdx.x;
  int beta = mat >> 6;
  stage_matrix(lA, T1 + mat * (PD * PD), threadIdx.x);     // [r][v]
  stage_matrix(lB, in2p + beta * (PD * PD), threadIdx.x);  // [v][s]
  stage_fence();

  int wave = threadIdx.x >> 5;
  int lane = threadIdx.x & 31;
  int half = lane >> 4, l16 = lane & 15;
  int m0 = wave * 16;
  v8f acc0 = {}, acc1 = {}, acc2 = {}, acc3 = {};
#pragma unroll
  for (int k = 0; k < PD; k += 4) {
    int kk = k + half * 2;
    v2f a = *(const v2f*)(lA + (m0 + l16) * ST + kk);      // contiguous pair
    v2f b0, b1, b2, b3;
    b0.x = lB[kk * ST + 0 + l16];  b0.y = lB[(kk + 1) * ST + 0 + l16];
    b1.x = lB[kk * ST + 16 + l16]; b1.y = lB[(kk + 1) * ST + 16 + l16];
    b2.x = lB[kk * ST + 32 + l16]; b2.y = lB[(kk + 1) * ST + 32 + l16];
    b3.x = lB[kk * ST + 48 + l16]; b3.y = lB[(kk + 1) * ST + 48 + l16];
    acc0 = WMMA_F32(a, b0, acc0);
    acc1 = WMMA_F32(a, b1, acc1);
    acc2 = WMMA_F32(a, b2, acc2);
    acc3 = WMMA_F32(a, b3, acc3);
  }
  float* D = T2 + mat * (PD * PD);
#pragma unroll
  for (int r = 0; r < 8; ++r) {
    int row = (m0 + r + half * 8) * PD + l16;
    D[row + 0]  = acc0[r];
    D[row + 16] = acc1[r];
    D[row + 32] = acc2[r];
    D[row + 48] = acc3[r];
  }
}

// ---------------------------------------------------------------------------
// GEMM3:  out[beta,K][Q] = sum_{z=(r,s)} T2[(beta,K)][z] * Cw[Q][z]
// Rows = 512 (beta*64+K), cols = 64 (Q), inner dim 4096. Both operands are
// contiguous in z (b64 loads). One wave per 16x64 strip; prefetch one k-block
// ahead. Store drops the zero padding.
// ---------------------------------------------------------------------------
__global__ __launch_bounds__(512) void gemm3(const float* __restrict__ T2,
                                             const float* __restrict__ Cw,
                                             float* __restrict__ out) {
  int t    = blockIdx.x * 16 + (threadIdx.x >> 5);   // row-tile id [0,32)
  int lane = threadIdx.x & 31;
  int half = lane >> 4, l16 = lane & 15;
  int m0 = t * 16;                       // row in [0,512): beta*64+K
  const float* Arow = T2 + (m0 + l16) * (PD * PD);
  v8f acc0 = {}, acc1 = {}, acc2 = {}, acc3 = {};
  for (int kb = 0; kb < PD * PD; kb += 64) {
    if (kb + 64 < PD * PD) {
      __builtin_prefetch(Arow + kb + 64 + half * 32, 0, 1);
      __builtin_prefetch(Cw + l16 * (PD * PD) + kb + 64 + half * 32, 0, 1);
    }
#pragma unroll 4
    for (int k = kb; k < kb + 64; k += 4) {
      int kk = k + half * 2;
      v2f a  = *(const v2f*)(Arow + kk);
      v2f b0 = *(const v2f*)(Cw + (0 + l16) * (PD * PD) + kk);
      v2f b1 = *(const v2f*)(Cw + (16 + l16) * (PD * PD) + kk);
      v2f b2 = *(const v2f*)(Cw + (32 + l16) * (PD * PD) + kk);
      v2f b3 = *(const v2f*)(Cw + (48 + l16) * (PD * PD) + kk);
      acc0 = WMMA_F32(a, b0, acc0);
      acc1 = WMMA_F32(a, b1, acc1);
      acc2 = WMMA_F32(a, b2, acc2);
      acc3 = WMMA_F32(a, b3, acc3);
    }
  }
#pragma unroll
  for (int r = 0; r < 8; ++r) {
    int row  = m0 + r + half * 8;        // beta*64 + K
    int beta = row >> 6;
    int K    = row & 63;
    if (K >= P) continue;
    float* orow = out + (beta * P + K) * P;
    if (0 + l16 < P)  orow[0 + l16]  = acc0[r];
    if (16 + l16 < P) orow[16 + l16] = acc1[r];
    if (32 + l16 < P) orow[32 + l16] = acc2[r];
    if (48 + l16 < P) orow[48 + l16] = acc3[r];
  }
}

// ---------------------------------------------------------------------------
extern "C" void kernel_launch(void* const* d_in, const int* in_sizes, int n_in,
                              void* d_out, int out_size, void* d_ws, size_t ws_size,
                              hipStream_t stream) {
  const float* in1    = (const float*)d_in[0];   // (B,S,2,S,2)
  const float* in2    = (const float*)d_in[1];   // (B,S,2,S,2)
  const float* weight = (const float*)d_in[2];   // (S,S)
  const float* F      = (const float*)d_in[3];   // (S,S,S,4)
  const float* T      = (const float*)d_in[4];   // (4,2,2,2)

  float* ws   = (float*)d_ws;
  float* Cw   = ws;                              //  64^3           = 262144
  float* in1p = Cw + PD * PD * PD;               //  8*64*64        =  32768
  float* in2p = in1p + NB * PD * PD;             //  8*64*64        =  32768
  float* T1   = in2p + NB * PD * PD;             //  512*64*64      = 2097152
  float* T2   = T1 + NB * PD * PD * PD;          //  512*64*64      = 2097152
  (void)in_sizes; (void)n_in; (void)out_size; (void)ws_size;

  build_coeff<<<(PD * PD * PD) / 256, 256, 0, stream>>>(weight, F, T, Cw);
  pad_inputs<<<(NB * PD * PD) / 256, 256, 0, stream>>>(in1, in2, in1p, in2p);
  gemm1<<<NB * PD, 128, 0, stream>>>(in1p, Cw, T1);
  gemm2<<<NB * PD, 128, 0, stream>>>(T1, in2p, T2);
  gemm3<<<2, 512, 0, stream>>>(T2, Cw, (float*)d_out);
}